// DensityAwareChamferDistance_65781719105608
// MI455X (gfx1250) — compile-verified
//
#include <hip/hip_runtime.h>
#include <math.h>

// Density-Aware Chamfer Distance for MI455X (gfx1250, wave32).
// NN search via V_WMMA_F32_16X16X4_F32 with the 4-dim distance embedding:
//   A row (target)  = ( tx, ty, tz, |t|^2 )
//   B col (query)   = ( -2qx, -2qy, -2qz, 1 )
//   D[m][n]         = -2 t.q + |t|^2      (add |q|^2 after min — monotone)

typedef __attribute__((ext_vector_type(2))) float v2f;
typedef __attribute__((ext_vector_type(8))) float v8f;

#define ALPHA    1000.0f
#define BATCH    4
#define CHUNK    4096          // target points staged in LDS per pass (64 KB)
#define WAVES    8             // waves per block (256 threads)

__global__ __launch_bounds__(256) void dacd_nn_kernel(
    const float* __restrict__ xyz1, const float* __restrict__ xyz2,
    float* __restrict__ dist, int* __restrict__ idx, int N)
{
    const int tileGroups = (N / 16) / WAVES;

    int bid   = blockIdx.x;
    int dir   = bid / (BATCH * tileGroups);            // 0: q=xyz2,t=xyz1 ; 1: q=xyz1,t=xyz2
    int rem   = bid % (BATCH * tileGroups);
    int batch = rem / tileGroups;
    int tg    = rem % tileGroups;

    const float* __restrict__ q_cloud = (dir == 0) ? xyz2 : xyz1;
    const float* __restrict__ t_cloud = (dir == 0) ? xyz1 : xyz2;

    __shared__ float lds[CHUNK * 4];                   // (x,y,z,|t|^2) per target

    const int lane = threadIdx.x & 31;
    const int wave = threadIdx.x >> 5;
    const int n    = lane & 15;                        // query column within tile
    const int half = lane >> 4;                        // 0: rows 0..7 / K0,K1 ; 1: rows 8..15 / K2,K3

    // ---- per-lane query (B operand, fixed for whole loop) ----
    const int qg = (tg * WAVES + wave) * 16 + n;       // global query index
    const float* qp = q_cloud + ((size_t)batch * N + qg) * 3;
    float qx = qp[0], qy = qp[1], qz = qp[2];
    float qn2 = qx * qx + qy * qy + qz * qz;
    v2f bq;
    bq.x = half ? (-2.0f * qz) : (-2.0f * qx);         // K2 : K0
    bq.y = half ? 1.0f         : (-2.0f * qy);         // K3 : K1

    float best    = 3.0e38f;
    int   bestIdx = 0;

    for (int chunk = 0; chunk < N; chunk += CHUNK) {
        __syncthreads();
        // stage CHUNK targets into LDS as float4 (x,y,z,norm2)
        for (int i = threadIdx.x; i < CHUNK; i += 256) {
            const float* tp = t_cloud + ((size_t)batch * N + chunk + i) * 3;
            float tx = tp[0], ty = tp[1], tz = tp[2];
            lds[i * 4 + 0] = tx;
            lds[i * 4 + 1] = ty;
            lds[i * 4 + 2] = tz;
            lds[i * 4 + 3] = tx * tx + ty * ty + tz * tz;
        }
        __syncthreads();

        // A operand: lane reads float2 at point*16B + half*8B  ->  ds_load_b64
        const v2f* __restrict__ lbase = ((const v2f*)lds) + half;

        for (int tt = 0; tt < CHUNK / 16; ++tt) {
            int p = tt * 16 + n;                       // target row M = n within tile
            v2f a = lbase[p * 2];

            v8f c = {0.f, 0.f, 0.f, 0.f, 0.f, 0.f, 0.f, 0.f};
            // (neg_a, A, neg_b, B, c_mod, C, reuse_a, reuse_b)
            c = __builtin_amdgcn_wmma_f32_16x16x4_f32(
                    false, a, false, bq, (short)0, c, false, false);

            const int base_idx = chunk + tt * 16 + half * 8;
#pragma unroll
            for (int r = 0; r < 8; ++r) {
                float v  = c[r];
                int   ti = base_idx + r;
                bool better = v < best;                // strict < keeps first index on ties
                bestIdx = better ? ti : bestIdx;
                best    = better ? v  : best;
            }
        }
    }

    // combine the two half-wave row groups (lane ^ 16 has same query, other rows)
    float ov = __shfl_xor(best, 16, 32);
    int   oi = __shfl_xor(bestIdx, 16, 32);
    if (ov < best || (ov == best && oi < bestIdx)) { best = ov; bestIdx = oi; }

    if (half == 0) {
        size_t og = ((size_t)dir * BATCH + batch) * N + qg;
        dist[og] = best + qn2;                         // true squared distance
        idx[og]  = bestIdx;
    }
}

__global__ void dacd_zero_kernel(int* __restrict__ counts, int total)
{
    int i = blockIdx.x * 256 + threadIdx.x;
    if (i < total) counts[i] = 0;
}

__global__ void dacd_count_kernel(const int* __restrict__ idx,
                                  int* __restrict__ counts, int N)
{
    int g  = blockIdx.x * 256 + threadIdx.x;           // over 2*B*N
    int bd = g / N;                                    // dir*B + batch
    atomicAdd(&counts[(size_t)bd * N + idx[g]], 1);
}

__global__ __launch_bounds__(256) void dacd_term_kernel(
    const float* __restrict__ dist, const int* __restrict__ idx,
    const int* __restrict__ counts, float* __restrict__ partials,
    int N, float frac21)
{
    int g   = blockIdx.x * 256 + threadIdx.x;
    int dir = g / (BATCH * N);
    int bd  = g / N;

    float d   = dist[g];
    int   cnt = counts[(size_t)bd * N + idx[g]];
    float w   = (float)cnt;                            // N_LAMBDA == 1.0

    float weight;
    if (dir == 0) weight = 1.0f / fmaxf(frac21 / w + 1e-6f, 1.0f);
    else          weight = 1.0f / (ceilf(frac21) * w + 1e-6f);

    float term = 1.0f - expf(-d * ALPHA) * weight;

    __shared__ float red[256];
    red[threadIdx.x] = term;
    __syncthreads();
#pragma unroll
    for (int s = 128; s > 0; s >>= 1) {
        if (threadIdx.x < s) red[threadIdx.x] += red[threadIdx.x + s];
        __syncthreads();
    }
    if (threadIdx.x == 0) partials[blockIdx.x] = red[0];
}

__global__ void dacd_final_kernel(const float* __restrict__ partials, int nPart,
                                  float* __restrict__ out, float scale)
{
    if (threadIdx.x == 0) {
        float s = 0.0f;
        for (int i = 0; i < nPart; ++i) s += partials[i];   // fixed order: deterministic
        out[0] = s * scale;
    }
}

extern "C" void kernel_launch(void* const* d_in, const int* in_sizes, int n_in,
                              void* d_out, int out_size, void* d_ws, size_t ws_size,
                              hipStream_t stream)
{
    const float* xyz1 = (const float*)d_in[0];         // prediction [B, n_x, 3]
    const float* xyz2 = (const float*)d_in[1];         // ground truth [B, n_gt, 3]

    const int n_x  = in_sizes[0] / (BATCH * 3);
    const int n_gt = in_sizes[1] / (BATCH * 3);
    const int N    = n_x;                              // reference: n_x == n_gt == 8192
    const float frac21 = (float)n_gt / (float)n_x;

    const int total = 2 * BATCH * N;                   // both directions

    // workspace layout
    float* dist     = (float*)d_ws;                    // total floats
    int*   idx      = (int*)(dist + total);            // total ints
    int*   counts   = idx + total;                     // total ints
    float* partials = (float*)(counts + total);        // total/256 floats

    const int tileGroups = (N / 16) / WAVES;
    const int nnBlocks   = 2 * BATCH * tileGroups;
    const int nTermBlk   = total / 256;

    dacd_nn_kernel<<<nnBlocks, 256, 0, stream>>>(xyz1, xyz2, dist, idx, N);
    dacd_zero_kernel<<<(total + 255) / 256, 256, 0, stream>>>(counts, total);
    dacd_count_kernel<<<total / 256, 256, 0, stream>>>(idx, counts, N);
    dacd_term_kernel<<<nTermBlk, 256, 0, stream>>>(dist, idx, counts, partials, N, frac21);
    dacd_final_kernel<<<1, 32, 0, stream>>>(partials, nTermBlk, (float*)d_out,
                                            1.0f / (2.0f * BATCH * N));
}